// MATMeshSurface_25159918420168
// MI455X (gfx1250) — compile-verified
//
#include <hip/hip_runtime.h>
#include <math.h>

// ---------------------------------------------------------------------------
// MAT mesh-surface point generation (lines + faces) for MI455X (gfx1250).
//
// Store-bandwidth-bound fp32 kernel (~139 MB traffic, ~6us floor @ 23.3TB/s):
//   * 134.4 MB write-once output      -> non-temporal b64 stores
//   * 800 KB skeleton table           -> temporal L2-resident b128 gathers
//   * 4 MB index streams (read once)  -> non-temporal loads
//   * trig over 8 fixed angles        -> fp32 LUT (no f64, no cosf lowering)
// One thread per (line, angle) pair; one thread per face.
// ---------------------------------------------------------------------------

namespace {

constexpr int kNG     = 8;
constexpr int kND     = 5;
constexpr int kNLines = 200000;
constexpr int kNFaces = 200000;

typedef float v2f __attribute__((ext_vector_type(2)));
typedef float v4f __attribute__((ext_vector_type(4)));
typedef int   v2i __attribute__((ext_vector_type(2)));

struct F3 { float x, y, z; };

__device__ __forceinline__ F3 f3add(F3 a, F3 b) { return {a.x + b.x, a.y + b.y, a.z + b.z}; }
__device__ __forceinline__ F3 f3sub(F3 a, F3 b) { return {a.x - b.x, a.y - b.y, a.z - b.z}; }
__device__ __forceinline__ F3 f3scl(F3 a, float s) { return {a.x * s, a.y * s, a.z * s}; }
__device__ __forceinline__ float f3dot(F3 a, F3 b) { return a.x * b.x + a.y * b.y + a.z * b.z; }
__device__ __forceinline__ F3 f3cross(F3 a, F3 b) {
  return {a.y * b.z - a.z * b.y, a.z * b.x - a.x * b.z, a.x * b.y - a.y * b.x};
}
__device__ __forceinline__ F3 f3norm(F3 a) {
  float inv = 1.0f / sqrtf(f3dot(a, a));
  return f3scl(a, inv);
}

// fp32-rounded constants matching the reference exactly.
__device__ constexpr float kCosPhi = 6.123234e-17f;   // float32(cos(pi/2) in float64)
__device__ constexpr float kC90    = -4.3711388e-8f;  // cos(float32(pi/2))
__device__ constexpr float kS90    = 1.0f;            // sin(float32(pi/2))

// (cos, sin) of float32(2*pi*k/8), k = 0..7, evaluated in fp32 like jnp.cos/sin.
// Epsilon entries come from the fp32 rounding of the angle (e.g. sin(pi_f32)).
__device__ const v2f kTrigLut[kNG] = {
    { 1.0f,           0.0f          },
    { 0.70710677f,    0.70710677f   },
    {-4.3711388e-8f,  1.0f          },
    {-0.70710677f,    0.70710677f   },
    {-1.0f,          -8.742278e-8f  },
    {-0.70710677f,   -0.70710677f   },
    { 1.1924881e-8f, -1.0f          },
    { 0.70710677f,   -0.70710677f   },
};

__device__ __forceinline__ void cone_intersect(F3 va, float ra, F3 vb, F3 vc, F3 nrm,
                                               F3& ip, F3& sn) {
  F3 cab = f3sub(vb, va);
  F3 cac = f3sub(vc, va);
  F3 pab = f3add(va, f3scl(f3norm(cab), kCosPhi * ra));
  F3 pac = f3add(va, f3scl(f3norm(cac), kCosPhi * ra));
  float u = cab.x, v = cab.y, w = cab.z;
  const float c = kC90, s = kS90;
  float t = 1.0f - c;
  float d0 = (u * u + (v * v + w * w) * c) * nrm.x + (u * v * t - w * s) * nrm.y +
             (u * w * t + v * s) * nrm.z;
  float d1 = (u * v * t + w * s) * nrm.x + (v * v + (u * u + w * w) * c) * nrm.y +
             (v * w * t - u * s) * nrm.z;
  float d2 = (u * w * t - v * s) * nrm.x + (v * w * t + u * s) * nrm.y +
             (w * w + (u * u + v * v) * c) * nrm.z;
  F3 dir = {d0, d1, d2};
  float tt = f3dot(f3sub(pac, pab), cac) / f3dot(dir, cac);
  ip = f3add(pab, f3scl(dir, tt));
  F3 vap = f3sub(ip, va);
  float mag = sqrtf(ra * ra - f3dot(vap, vap));   // NaN when negative, same as jnp
  sn = f3scl(nrm, mag);
}

__device__ __forceinline__ void store18_nt(float* base, const float* p) {
  v2f* o = reinterpret_cast<v2f*>(base);          // base is 8-byte aligned by construction
#pragma unroll
  for (int j = 0; j < 9; ++j) {
    v2f v;
    v.x = p[2 * j];
    v.y = p[2 * j + 1];
    __builtin_nontemporal_store(v, o + j);
  }
}

}  // namespace

// -------------------------- lines: 1 thread per (line, angle) ---------------
__global__ __launch_bounds__(256) void mat_lines_kernel(const v4f* __restrict__ skel,
                                                        const v2i* __restrict__ lines,
                                                        float* __restrict__ out) {
  int tid = blockIdx.x * blockDim.x + threadIdx.x;
  if (tid >= kNLines * kNG) return;
  int line = tid >> 3;   // 8 consecutive lanes share one line -> broadcast gathers
  int g    = tid & 7;

  v2i ln = __builtin_nontemporal_load(&lines[line]);   // one b64 NT load per thread
  v4f a = skel[ln.x];    // global_load_b128, hot in the 192MB L2 (table = 800KB)
  v4f b = skel[ln.y];

  F3 v1 = {a.x, a.y, a.z}; float r1 = a.w;
  F3 v2 = {b.x, b.y, b.z}; float r2 = b.w;

  F3 c12  = f3sub(v2, v1);
  F3 perp = {1.0f, 1.0f, (-c12.x - c12.y) / c12.z};
  F3 d0   = f3norm(perp);
  F3 axis = f3norm(c12);

  v2f cs = kTrigLut[g];                                // one b64 load, 64B hot table
  float ca = cs.x;
  float sa = cs.y;

  F3 kxd = f3cross(axis, d0);
  F3 kdd = f3scl(axis, f3dot(axis, d0));
  F3 rot = f3add(f3add(f3scl(d0, ca), f3scl(kxd, sa)), f3scl(kdd, 1.0f - ca));
  F3 nor = f3norm(rot);

  F3 p1 = f3add(v1, f3scl(nor, r1));
  F3 p2 = f3add(v2, f3scl(nor, r2));
  F3 dp = f3sub(p2, p1);

  float pts[18];
  pts[0] = p1.x; pts[1] = p1.y; pts[2] = p1.z;
  pts[3] = p2.x; pts[4] = p2.y; pts[5] = p2.z;
#pragma unroll
  for (int k = 1; k < kND; ++k) {                 // fr = k/5, k = 1..4
    float fr = (float)k / (float)kND;
    pts[(k + 1) * 3 + 0] = p1.x + dp.x * fr;
    pts[(k + 1) * 3 + 1] = p1.y + dp.y * fr;
    pts[(k + 1) * 3 + 2] = p1.z + dp.z * fr;
  }

  // Flat layout: ((line*8 + g)*6 + d)*3 == tid*18 + d*3. 72B/thread, contiguous per wave.
  store18_nt(out + (size_t)tid * 18, pts);
}

// -------------------------- faces: 1 thread per face ------------------------
__global__ __launch_bounds__(256) void mat_faces_kernel(const v4f* __restrict__ skel,
                                                        const int* __restrict__ faces,
                                                        float* __restrict__ out) {
  int f = blockIdx.x * blockDim.x + threadIdx.x;
  if (f >= kNFaces) return;

  int i0 = __builtin_nontemporal_load(&faces[f * 3 + 0]);
  int i1 = __builtin_nontemporal_load(&faces[f * 3 + 1]);
  int i2 = __builtin_nontemporal_load(&faces[f * 3 + 2]);
  v4f A = skel[i0];
  v4f B = skel[i1];
  v4f C = skel[i2];

  F3 v1 = {A.x, A.y, A.z}; float r1 = A.w;
  F3 v2 = {B.x, B.y, B.z}; float r2 = B.w;
  F3 v3 = {C.x, C.y, C.z}; float r3 = C.w;

  F3 nrm = f3norm(f3cross(f3sub(v1, v2), f3sub(v1, v3)));

  F3 ip1, sn1, ip2, sn2, ip3, sn3;
  cone_intersect(v1, r1, v2, v3, nrm, ip1, sn1);
  cone_intersect(v2, r2, v1, v3, nrm, ip2, sn2);
  cone_intersect(v3, r3, v1, v2, nrm, ip3, sn3);

  F3 pf0 = f3add(ip1, sn1), pf1 = f3add(ip2, sn2), pf2 = f3add(ip3, sn3);
  F3 pf3 = f3sub(ip1, sn1), pf4 = f3sub(ip2, sn2), pf5 = f3sub(ip3, sn3);

  float pts[18] = {pf0.x, pf0.y, pf0.z, pf1.x, pf1.y, pf1.z, pf2.x, pf2.y, pf2.z,
                   pf3.x, pf3.y, pf3.z, pf4.x, pf4.y, pf4.z, pf5.x, pf5.y, pf5.z};
  store18_nt(out + (size_t)f * 18, pts);

  const float third = 1.0f / 3.0f;
  F3 bar0 = f3scl(f3add(f3add(pf0, pf1), pf2), third);
  F3 bar1 = f3scl(f3add(f3add(pf3, pf4), pf5), third);
  float bb[6] = {bar0.x, bar0.y, bar0.z, bar1.x, bar1.y, bar1.z};
  v2f* ob = reinterpret_cast<v2f*>(out + (size_t)kNFaces * 18 + (size_t)f * 6);
#pragma unroll
  for (int j = 0; j < 3; ++j) {
    v2f v;
    v.x = bb[2 * j];
    v.y = bb[2 * j + 1];
    __builtin_nontemporal_store(v, ob + j);
  }
}

// ---------------------------------------------------------------------------
extern "C" void kernel_launch(void* const* d_in, const int* in_sizes, int n_in,
                              void* d_out, int out_size, void* d_ws, size_t ws_size,
                              hipStream_t stream) {
  (void)in_sizes; (void)n_in; (void)out_size; (void)d_ws; (void)ws_size;

  const v4f* skel  = (const v4f*)d_in[0];   // (N_SKEL, 4) float32
  const v2i* lines = (const v2i*)d_in[1];   // (N_LINES, 2) int32
  const int* faces = (const int*)d_in[2];   // (N_FACES, 3) int32
  float* out = (float*)d_out;

  const int n_line_threads = kNLines * kNG;                 // 1.6M threads = 50K waves
  mat_lines_kernel<<<(n_line_threads + 255) / 256, 256, 0, stream>>>(skel, lines, out);

  float* out_faces = out + (size_t)kNLines * kNG * 6 * 3;   // faces region after lines
  mat_faces_kernel<<<(kNFaces + 255) / 256, 256, 0, stream>>>(skel, faces, out_faces);
}